// DecodeYoloV1_22694607192619
// MI455X (gfx1250) — compile-verified
//
#include <hip/hip_runtime.h>
#include <math.h>

// ---------------------------------------------------------------------------
// YOLOv1 decode + NMS for MI455X (gfx1250).
//
// Problem size: 1x7x7x30 f32 in (5.9 KB), 49x6 boxes + 49 keep out (1.4 KB).
// At 23.3 TB/s HBM this is ~0.3 ns of bandwidth -> entirely latency bound.
// Optimal shape: ONE workgroup, 64 threads (2 wave32), all state in LDS.
// CDNA5-specific path used: async global->LDS DMA (ASYNCcnt) to issue the
// whole 5.9 KB fetch up front, drained with s_wait_asynccnt before compute.
// ---------------------------------------------------------------------------

#define S7        7
#define NCELL     49          // 7*7
#define NCH       30          // 2*5 + 20
#define NCLS      20
#define CONF_THR  0.25f
#define IOU_THR   0.45f
#define EPSV      1e-6f
#define STRIDE_PX 64.0f       // 448 / 7

// Vector pointee types required by the async-to-LDS builtins (from clang's
// diagnostic: b128 takes v4i pointers, b64 takes v2i pointers).
typedef int vi4 __attribute__((__vector_size__(16)));
typedef int vi2 __attribute__((__vector_size__(8)));

// ---- probe CDNA5 async-to-LDS builtins (compile-safe fallback) ------------
#if defined(__has_builtin)
#  if __has_builtin(__builtin_amdgcn_global_load_async_to_lds_b128) && \
      __has_builtin(__builtin_amdgcn_global_load_async_to_lds_b64)
#    define HAVE_ASYNC_LDS 1
#  endif
#  if __has_builtin(__builtin_amdgcn_s_wait_asynccnt)
#    define WAIT_ASYNC0() __builtin_amdgcn_s_wait_asynccnt(0)
#  endif
#endif
#ifndef HAVE_ASYNC_LDS
#  define HAVE_ASYNC_LDS 0
#  pragma message("probe: __builtin_amdgcn_global_load_async_to_lds_* NOT available; plain-load fallback in use")
#endif
#ifndef WAIT_ASYNC0
#  define WAIT_ASYNC0() asm volatile("s_wait_asynccnt 0" ::: "memory")
#endif

__device__ __forceinline__ float sigm(float v) {
    return 1.0f / (1.0f + __expf(-v));
}

__global__ __launch_bounds__(64) void yolo_decode_nms_kernel(
    const float* __restrict__ x, float* __restrict__ out)
{
    __shared__ __align__(16) float sx[NCELL * NCH];  // raw logits (1470 f32)
    __shared__ float dec[NCELL][6];     // decoded: cls, conf, cx, cy, w, h
    __shared__ float srt[NCELL][6];     // sorted by conf desc (stable)
    __shared__ int   keep_s[NCELL];

    const int tid = threadIdx.x;

    // ---------------- stage input: global -> LDS -------------------------
#if HAVE_ASYNC_LDS
    // 367 x B128 chunks cover floats [0, 1468); one B64 covers [1468, 1470).
    for (int c = tid; c < 367; c += 64) {
        __builtin_amdgcn_global_load_async_to_lds_b128(
            (__attribute__((address_space(1))) vi4*)(x + 4 * c),
            (__attribute__((address_space(3))) vi4*)(&sx[4 * c]),
            /*offset=*/0, /*cpol=*/0);
    }
    if (tid == 0) {
        __builtin_amdgcn_global_load_async_to_lds_b64(
            (__attribute__((address_space(1))) vi2*)(x + 1468),
            (__attribute__((address_space(3))) vi2*)(&sx[1468]),
            /*offset=*/0, /*cpol=*/0);
    }
    WAIT_ASYNC0();             // drain this wave's ASYNCcnt
#else
    for (int i = tid; i < NCELL * NCH; i += 64) sx[i] = x[i];
#endif
    __syncthreads();           // LDS now visible to both waves

    // ---------------- decode: one thread per cell ------------------------
    if (tid < NCELL) {
        const int gy = tid / S7;
        const int gx = tid - gy * S7;
        const float* cell = &sx[tid * NCH];

        // class argmax over sigmoid(logits[0:20]); strict > -> first max wins
        float bestv = sigm(cell[0]);
        int   bidx  = 0;
        for (int j = 1; j < NCLS; ++j) {
            float v = sigm(cell[j]);
            if (v > bestv) { bestv = v; bidx = j; }
        }

        // best-of-2 box by objectness; strict > -> box 0 wins ties (argmax)
        const float c0  = sigm(cell[NCLS + 0]);
        const float c1  = sigm(cell[NCLS + 5]);
        const int   sel = (c1 > c0) ? 1 : 0;
        const float pconf = sel ? c1 : c0;
        const int   bb  = NCLS + 1 + 5 * sel;

        const float bx = sigm(cell[bb + 0]);
        const float by = sigm(cell[bb + 1]);
        const float bw = sigm(cell[bb + 2]);
        const float bh = sigm(cell[bb + 3]);

        dec[tid][0] = (float)bidx;
        dec[tid][1] = pconf;
        dec[tid][2] = (bx + (float)gx) * STRIDE_PX;
        dec[tid][3] = (by + (float)gy) * STRIDE_PX;
        dec[tid][4] = bw * (float)S7 * STRIDE_PX;   // bw * 7 * 64
        dec[tid][5] = bh * (float)S7 * STRIDE_PX;
    }
    __syncthreads();

    // ---------------- stable sort by -conf via O(N^2) rank ----------------
    if (tid < NCELL) {
        const float ci = dec[tid][1];
        int rank = 0;
        for (int j = 0; j < NCELL; ++j) {
            const float cj = dec[j][1];
            rank += (cj > ci) || (cj == ci && j < tid);
        }
        #pragma unroll
        for (int k = 0; k < 6; ++k) srt[rank][k] = dec[tid][k];
    }
    __syncthreads();

    if (tid < NCELL) keep_s[tid] = (srt[tid][1] > CONF_THR) ? 1 : 0;
    __syncthreads();

    // ---------------- sequential NMS (49 rounds) --------------------------
    for (int i = 0; i < NCELL; ++i) {
        if (tid < NCELL && tid > i) {
            // live keep[i] semantics: clearing an already-0 bit is a no-op,
            // so gating on keep_s[i] alone matches keep & ~suppress.
            if (keep_s[i] && keep_s[tid] && (srt[tid][0] == srt[i][0])) {
                const float iw2 = srt[i][4] * 0.5f, ih2 = srt[i][5] * 0.5f;
                const float tw2 = srt[tid][4] * 0.5f, th2 = srt[tid][5] * 0.5f;
                const float b1minx = srt[i][2] - iw2, b1maxx = srt[i][2] + iw2;
                const float b1miny = srt[i][3] - ih2, b1maxy = srt[i][3] + ih2;
                const float b2minx = srt[tid][2] - tw2, b2maxx = srt[tid][2] + tw2;
                const float b2miny = srt[tid][3] - th2, b2maxy = srt[tid][3] + th2;

                const float interw = fmaxf(fminf(b1maxx, b2maxx) - fmaxf(b1minx, b2minx), 0.0f);
                const float interh = fmaxf(fminf(b1maxy, b2maxy) - fmaxf(b1miny, b2miny), 0.0f);
                const float inter  = interw * interh;
                const float a1 = fabsf((b1maxx - b1minx) * (b1maxy - b1miny));
                const float a2 = fabsf((b2maxx - b2minx) * (b2maxy - b2miny));
                const float iou = inter / (a1 + a2 - inter + EPSV);
                if (iou >= IOU_THR) keep_s[tid] = 0;
            }
        }
        __syncthreads();
    }

    // ---------------- write outputs ---------------------------------------
    // d_out layout: boxes[49][6] (b * keep) followed by keep[49] as 0/1.
    if (tid < NCELL) {
        const float kf = keep_s[tid] ? 1.0f : 0.0f;
        #pragma unroll
        for (int k = 0; k < 6; ++k) out[tid * 6 + k] = srt[tid][k] * kf;
        out[NCELL * 6 + tid] = kf;
    }
}

extern "C" void kernel_launch(void* const* d_in, const int* in_sizes, int n_in,
                              void* d_out, int out_size, void* d_ws, size_t ws_size,
                              hipStream_t stream) {
    (void)in_sizes; (void)n_in; (void)out_size; (void)d_ws; (void)ws_size;
    const float* x = (const float*)d_in[0];
    float* out = (float*)d_out;
    // One workgroup of 64 threads (2 wave32) owns the whole problem:
    // the workload is launch-latency bound, not bandwidth/compute bound.
    yolo_decode_nms_kernel<<<1, 64, 0, stream>>>(x, out);
}